// ConvFilter_4054449127838
// MI455X (gfx1250) — compile-verified
//
#include <hip/hip_runtime.h>
#include <hip/hip_bf16.h>
#include <math.h>

// Problem constants (from reference)
#define BB    32
#define LL    2048
#define FF    512
#define KK    128
#define LOUTC 1921            // LL - KK + 1
#define NCHUNK 8
#define TCHUNK ((LOUTC + NCHUNK - 1) / NCHUNK)   // 241

typedef __attribute__((ext_vector_type(2))) float v2f;
typedef __attribute__((ext_vector_type(8))) float v8f;

// ---------------------------------------------------------------------------
// Kernel 1: tau = sigmoid((feat @ W^T + b)/10 - 3) via V_WMMA_F32_16X16X4_F32
// feat[b,k] = x[b, L-1, k];  acc[m,n] = sum_k feat[m,k] * W[n,k]
// One wave (32 lanes) computes a 16x16 tile of the [32 x 512] result.
// Epilogue: tau, tau^128 (7 squarings), norm = (1-tau)/(1-tau^128).
// ---------------------------------------------------------------------------
__global__ __launch_bounds__(32)
void tau_gemm_wmma(const float* __restrict__ x,
                   const float* __restrict__ W,
                   const float* __restrict__ bias,
                   float* __restrict__ tau_out,
                   float* __restrict__ norm_out,
                   float* __restrict__ c128_out)
{
    const int lane = threadIdx.x & 31;
    const int half = lane >> 4;      // 0: lanes 0-15, 1: lanes 16-31
    const int l    = lane & 15;

    const int n0 = blockIdx.x * 16;  // N tile (0..511)
    const int m0 = blockIdx.y * 16;  // M tile (0..31)

    // A row pointer: feat[m0+l, :] = x[(m0+l)*L*F + (L-1)*F + ...]
    const float* arow = x + (size_t)(m0 + l) * (size_t)(LL * FF) + (size_t)(LL - 1) * FF;
    // B column n = W row (n0+l): B[k, n] = W[n, k]
    const float* brow = W + (size_t)(n0 + l) * FF;

    v8f acc = {};

    // K = 512 in steps of 4. A frag: lane holds A[M=l, K=k+2*half .. +1]
    //                         B frag: lane holds B[K=k+2*half .. +1, N=n0+l]
    for (int k = 0; k < FF; k += 4) {
        v2f a, b;
        a.x = arow[k + 2 * half + 0];
        a.y = arow[k + 2 * half + 1];
        b.x = brow[k + 2 * half + 0];
        b.y = brow[k + 2 * half + 1];
        acc = __builtin_amdgcn_wmma_f32_16x16x4_f32(
            /*neg_a=*/false, a, /*neg_b=*/false, b,
            /*c_mod=*/(short)0, acc, /*reuse_a=*/false, /*reuse_b=*/false);
    }

    // Epilogue: C VGPR r holds [M = m0 + 8*half + r, N = n0 + l]
    const int n = n0 + l;
    const float bn = bias[n];
#pragma unroll
    for (int r = 0; r < 8; ++r) {
        const int m = m0 + 8 * half + r;
        float z   = (acc[r] + bn) * 0.1f - 3.0f;
        float tau = 1.0f / (1.0f + __expf(-z));     // in (0,1)
        // tau^128 via 7 squarings
        float p = tau;
        p = p * p;  // ^2
        p = p * p;  // ^4
        p = p * p;  // ^8
        p = p * p;  // ^16
        p = p * p;  // ^32
        p = p * p;  // ^64
        p = p * p;  // ^128
        float om    = 1.0f - tau;
        float denom = 1.0f - p;
        float nrm   = (om > 1e-12f && denom > 1e-12f) ? (om / denom) : (1.0f / (float)KK);
        const size_t idx = (size_t)m * FF + n;
        tau_out[idx]  = tau;
        norm_out[idx] = nrm;
        c128_out[idx] = p;   // tau^K
    }
}

// ---------------------------------------------------------------------------
// Kernel 2: exponential-decay FIR via exact sliding recurrence.
//   S[t] = sum_{i=0}^{K-1} tau^{K-1-i} x[t+i]
//   S[t] = tau*S[t-1] + x[t+K-1] - tau^K * x[t-1];  out[t] = norm * S[t]
// Thread = one (b, f) pair over one t-chunk; warmup rebuilds S at chunk start
// (Horner, 128 FMAs). Coalesced across f (lane stride 4B). Streams stay in L2.
// ---------------------------------------------------------------------------
__global__ __launch_bounds__(256)
void fir_decay_kernel(const float* __restrict__ x,
                      const float* __restrict__ tau_a,
                      const float* __restrict__ norm_a,
                      const float* __restrict__ c128_a,
                      float* __restrict__ out)
{
    const int f     = blockIdx.x * blockDim.x + threadIdx.x;  // 0..511
    const int chunk = blockIdx.y;                              // 0..NCHUNK-1
    const int b     = blockIdx.z;                              // 0..31

    int t0 = chunk * TCHUNK;
    int t1 = t0 + TCHUNK;
    if (t1 > LOUTC) t1 = LOUTC;
    if (t0 >= t1) return;

    const size_t pidx = (size_t)b * FF + f;
    const float tau = tau_a[pidx];
    const float nrm = norm_a[pidx];
    const float c   = c128_a[pidx];     // tau^K

    const float* xb = x   + (size_t)b * (size_t)(LL * FF) + f;
    float*       ob = out + (size_t)b * (size_t)(LOUTC)*FF + f;

    // Warmup: S = sum_{i=0}^{K-1} tau^{K-1-i} x[t0+i]  (Horner over ascending i)
    float S = 0.0f;
#pragma unroll 8
    for (int i = 0; i < KK; ++i) {
        S = fmaf(S, tau, xb[(size_t)(t0 + i) * FF]);
    }
    ob[(size_t)t0 * FF] = nrm * S;

    // Steady state: 2 loads + 2 FMA + 1 mul + 1 store per output
    for (int t = t0 + 1; t < t1; ++t) {
        float xn = xb[(size_t)(t + KK - 1) * FF];
        float xo = xb[(size_t)(t - 1) * FF];
        S = fmaf(tau, S, fmaf(-c, xo, xn));
        ob[(size_t)t * FF] = nrm * S;
    }
}

// ---------------------------------------------------------------------------
extern "C" void kernel_launch(void* const* d_in, const int* in_sizes, int n_in,
                              void* d_out, int out_size, void* d_ws, size_t ws_size,
                              hipStream_t stream)
{
    const float* x  = (const float*)d_in[0];   // [B, L, F]
    const float* W  = (const float*)d_in[1];   // [F, F]
    const float* bv = (const float*)d_in[2];   // [F]
    float* out = (float*)d_out;                // [B, LOUT, F]

    float* ws   = (float*)d_ws;
    float* tauA = ws;                          // B*F
    float* nrmA = ws + (size_t)BB * FF;        // B*F
    float* cA   = ws + (size_t)2 * BB * FF;    // B*F

    // Kernel 1: 32 N-tiles x 2 M-tiles, one wave each
    dim3 g1(FF / 16, BB / 16, 1);
    tau_gemm_wmma<<<g1, dim3(32, 1, 1), 0, stream>>>(x, W, bv, tauA, nrmA, cA);

    // Kernel 2: grid (F/256, NCHUNK, B), block 256
    dim3 g2(FF / 256, NCHUNK, BB);
    fir_decay_kernel<<<g2, dim3(256, 1, 1), 0, stream>>>(x, tauA, nrmA, cA, out);
}